// ResidualGNNLayer_36919538876531
// MI455X (gfx1250) — compile-verified
//
#include <hip/hip_runtime.h>
#include <hip/hip_bf16.h>

#define D 64
#define LN_EPS 1e-5f

typedef __attribute__((ext_vector_type(2))) float v2f;
typedef __attribute__((ext_vector_type(8))) float v8f;

// ---------------------------------------------------------------------------
// Kernel 1: initialize workspace. deg[i] = 1.0 (self loop), agg[:] = 0.
// ---------------------------------------------------------------------------
__global__ void init_ws_kernel(float* __restrict__ deg, float* __restrict__ agg,
                               int n_nodes, int n_feat_total) {
    int i = blockIdx.x * blockDim.x + threadIdx.x;
    if (i < n_feat_total) agg[i] = 0.0f;
    if (i < n_nodes) deg[i] = 1.0f;
}

// ---------------------------------------------------------------------------
// Kernel 2: in-degree via atomics (self loop already baked in as 1.0).
// ---------------------------------------------------------------------------
__global__ void degree_kernel(const int* __restrict__ dst, float* __restrict__ deg,
                              int n_edges) {
    int e = blockIdx.x * blockDim.x + threadIdx.x;
    if (e < n_edges) atomicAdd(&deg[dst[e]], 1.0f);
}

// ---------------------------------------------------------------------------
// Kernel 3: dinv = rsqrt(deg). deg >= 1 always (self loop), so no zero guard.
// ---------------------------------------------------------------------------
__global__ void rsqrt_kernel(const float* __restrict__ deg, float* __restrict__ dinv,
                             int n_nodes) {
    int i = blockIdx.x * blockDim.x + threadIdx.x;
    if (i < n_nodes) dinv[i] = rsqrtf(deg[i]);
}

// ---------------------------------------------------------------------------
// Kernel 4: h = x @ W with V_WMMA_F32_16X16X4_F32 (full f32 precision).
// One wave32 per 16x16 output tile. blockDim = 128 (4 waves); wave w owns
// column tile n0 = 16*w of the same 16-row stripe. K = 64 -> 16 WMMA steps.
//
// VGPR layouts (ISA 7.12.2, wave32):
//   A (16x4, MxK):  lane l: M = l%16, half = l/16; VGPR0 = K(2*half),
//                   VGPR1 = K(2*half+1)
//   B (4x16, KxN):  lane l: N = l%16, half = l/16; VGPR0 = row K(2*half),
//                   VGPR1 = row K(2*half+1)
//   C/D (16x16):    lane l: N = l%16; VGPR v: M = v + 8*(l/16)
// EXEC must be all 1s: grid covers only full 16-row tiles, no divergence.
// ---------------------------------------------------------------------------
__global__ void gemm_wmma_kernel(const float* __restrict__ x,
                                 const float* __restrict__ W,
                                 float* __restrict__ h) {
    const int lane = threadIdx.x & 31;
    const int wave = threadIdx.x >> 5;      // 0..3 -> N tile
    const int m0   = blockIdx.x * 16;
    const int n0   = wave * 16;
    const int half = lane >> 4;             // 0 or 1
    const int ml   = lane & 15;             // M for A, N for B/C/D

    const float* __restrict__ xrow = x + (size_t)(m0 + ml) * D;

    v8f acc = {};
#pragma unroll
    for (int k0 = 0; k0 < D; k0 += 4) {
        const int k = k0 + 2 * half;
        v2f a;
        a.x = xrow[k];
        a.y = xrow[k + 1];
        v2f b;
        b.x = W[(size_t)k * D + n0 + ml];
        b.y = W[(size_t)(k + 1) * D + n0 + ml];
        acc = __builtin_amdgcn_wmma_f32_16x16x4_f32(
            /*neg_a=*/false, a, /*neg_b=*/false, b,
            /*c_mod=*/(short)0, acc, /*reuse_a=*/false, /*reuse_b=*/false);
    }

    float* __restrict__ hout = h + n0 + ml;
#pragma unroll
    for (int v = 0; v < 8; ++v) {
        hout[(size_t)(m0 + v + 8 * half) * D] = acc[v];
    }
}

// Scalar tail for rows not covered by full 16-row tiles (unused when N%16==0).
__global__ void gemm_tail_kernel(const float* __restrict__ x,
                                 const float* __restrict__ W,
                                 float* __restrict__ h,
                                 int row_start, int n_nodes) {
    int gid = blockIdx.x * blockDim.x + threadIdx.x;
    int row = row_start + (gid >> 6);
    int col = gid & (D - 1);
    if (row >= n_nodes) return;
    float acc = 0.0f;
    const float* xr = x + (size_t)row * D;
#pragma unroll 8
    for (int k = 0; k < D; ++k) acc += xr[k] * W[(size_t)k * D + col];
    h[(size_t)row * D + col] = acc;
}

// ---------------------------------------------------------------------------
// Kernel 5: edge scatter. 64 threads per edge (one per feature):
//   agg[dst] += h[src] * dinv[src] * dinv[dst]
// ---------------------------------------------------------------------------
__global__ void edge_scatter_kernel(const int* __restrict__ src,
                                    const int* __restrict__ dst,
                                    const float* __restrict__ h,
                                    const float* __restrict__ dinv,
                                    float* __restrict__ agg,
                                    int n_edges) {
    int gid = blockIdx.x * blockDim.x + threadIdx.x;
    int e = gid >> 6;           // D == 64 features per edge
    int j = gid & (D - 1);
    if (e >= n_edges) return;
    int s = src[e];
    int d = dst[e];
    float w = dinv[s] * dinv[d];
    atomicAdd(&agg[(size_t)d * D + j], h[(size_t)s * D + j] * w);
}

// ---------------------------------------------------------------------------
// Kernel 6: finalize. One wave32 per node, 2 features per lane.
//   y = x + agg + h*dinv^2 (self-loop term) + b;  out = LN(y)*gamma + beta
// ---------------------------------------------------------------------------
__global__ void finalize_kernel(const float* __restrict__ x,
                                const float* __restrict__ h,
                                const float* __restrict__ agg,
                                const float* __restrict__ dinv,
                                const float* __restrict__ bias,
                                const float* __restrict__ gamma,
                                const float* __restrict__ beta,
                                float* __restrict__ out,
                                int n_nodes) {
    int node = blockIdx.x * (blockDim.x >> 5) + (threadIdx.x >> 5);
    if (node >= n_nodes) return;
    int lane = threadIdx.x & 31;

    float di = dinv[node];
    float selfw = di * di;                  // dinv[i]*dinv[i] self-loop norm

    size_t base = (size_t)node * D + lane * 2;
    float2 xv = *(const float2*)(x + base);
    float2 hv = *(const float2*)(h + base);
    float2 av = *(const float2*)(agg + base);
    float2 bv = *(const float2*)(bias + lane * 2);

    float y0 = xv.x + av.x + hv.x * selfw + bv.x;
    float y1 = xv.y + av.y + hv.y * selfw + bv.y;

    // wave32 reduction for sum and sum-of-squares over 64 features
    float s  = y0 + y1;
    float sq = y0 * y0 + y1 * y1;
#pragma unroll
    for (int off = 16; off > 0; off >>= 1) {
        s  += __shfl_xor(s, off, 32);
        sq += __shfl_xor(sq, off, 32);
    }
    float mu   = s * (1.0f / D);
    float var  = sq * (1.0f / D) - mu * mu;
    float rstd = rsqrtf(var + LN_EPS);

    float2 gv = *(const float2*)(gamma + lane * 2);
    float2 be = *(const float2*)(beta + lane * 2);
    float2 o;
    o.x = (y0 - mu) * rstd * gv.x + be.x;
    o.y = (y1 - mu) * rstd * gv.y + be.y;
    *(float2*)(out + base) = o;
}

// ---------------------------------------------------------------------------
// Launcher
// ---------------------------------------------------------------------------
extern "C" void kernel_launch(void* const* d_in, const int* in_sizes, int n_in,
                              void* d_out, int out_size, void* d_ws, size_t ws_size,
                              hipStream_t stream) {
    const float* x     = (const float*)d_in[0];
    const float* W     = (const float*)d_in[1];
    const float* bias  = (const float*)d_in[2];
    const float* gamma = (const float*)d_in[3];
    const float* beta  = (const float*)d_in[4];
    const int*   edge  = (const int*)d_in[5];

    const int n_nodes = in_sizes[0] / D;
    const int n_edges = in_sizes[5] / 2;
    const int* src = edge;
    const int* dst = edge + n_edges;

    // Workspace layout (floats, 1KB-aligned stripes):
    //   deg[NP], dinv[NP], h[NP*D], agg[NP*D]
    size_t NP = ((size_t)n_nodes + 255) & ~(size_t)255;
    float* ws   = (float*)d_ws;
    float* deg  = ws;
    float* dinv = ws + NP;
    float* h    = ws + 2 * NP;
    float* agg  = h + NP * (size_t)D;

    float* out = (float*)d_out;

    const int nfeat = n_nodes * D;

    // 1) init workspace
    init_ws_kernel<<<(nfeat + 255) / 256, 256, 0, stream>>>(deg, agg, n_nodes, nfeat);

    // 2) degree
    degree_kernel<<<(n_edges + 255) / 256, 256, 0, stream>>>(dst, deg, n_edges);

    // 3) dinv
    rsqrt_kernel<<<(n_nodes + 255) / 256, 256, 0, stream>>>(deg, dinv, n_nodes);

    // 4) h = x @ W  (WMMA f32; independent of 2/3, just queued in order)
    int full_tiles = n_nodes / 16;
    if (full_tiles > 0)
        gemm_wmma_kernel<<<full_tiles, 128, 0, stream>>>(x, W, h);
    int tail_rows = n_nodes - full_tiles * 16;
    if (tail_rows > 0)
        gemm_tail_kernel<<<(tail_rows * D + 255) / 256, 256, 0, stream>>>(
            x, W, h, full_tiles * 16, n_nodes);

    // 5) edge messages + scatter-add
    long long edge_threads = (long long)n_edges * D;
    edge_scatter_kernel<<<(int)((edge_threads + 255) / 256), 256, 0, stream>>>(
        src, dst, h, dinv, agg, n_edges);

    // 6) residual + bias + self-loop + LayerNorm
    finalize_kernel<<<(n_nodes + 7) / 8, 256, 0, stream>>>(
        x, h, agg, dinv, bias, gamma, beta, out, n_nodes);
}